// Q_Model_16999480557726
// MI455X (gfx1250) — compile-verified
//
#include <hip/hip_runtime.h>
#include <cstdint>

// Quantum circuit factorizes per-wire (no entangling gates):
//   v_w = M2 * RY(x) * M1 * RY(x) * M0 * RY(x) * |0>,  z_w = |v0|^2 - |v1|^2
// M_l = RZ(p2) RY(p1) RZ(p0) are 2x2 complex, shared across pixels.
//
// Roofline: 12 MB traffic (~0.5 us @ 23.3 TB/s) vs per-pixel VALU. With
// hardware v_sin/v_cos for the 3 per-pixel encoder angles the VALU term
// drops below the HBM term -> memory-bound, which is the floor for this op.

#define THREADS 256

// ---------------------------------------------------------------------------
// Kernel A (runs once, 9 threads): compose the per-(wire,layer) 2x2 complex
// gates from the 27 params, using fully accurate sincosf.
// Layout in ws: (w*3 + l)*8 floats = [m00r,m00i, m01r,m01i, m10r,m10i, m11r,m11i]
//   M = RZ(p2) RY(p1) RZ(p0):
//   M00 =  c1 e^{-iA},  M01 = -s1 e^{+iB},  M10 = s1 e^{-iB},  M11 = c1 e^{+iA}
//   A = (p0+p2)/2, B = (p0-p2)/2, c1/s1 = cos/sin(p1/2)
// ---------------------------------------------------------------------------
__global__ void compose_gates(const float* __restrict__ params,
                              float* __restrict__ mats) {
    int i = threadIdx.x;
    if (i >= 9) return;
    int l = i / 3, w = i % 3;
    float p0 = params[l * 9 + 3 * w + 0];
    float p1 = params[l * 9 + 3 * w + 1];
    float p2 = params[l * 9 + 3 * w + 2];
    float cA, sA, cB, sB, c1, s1;
    sincosf(0.5f * (p0 + p2), &sA, &cA);
    sincosf(0.5f * (p0 - p2), &sB, &cB);
    sincosf(0.5f * p1,        &s1, &c1);
    float* m = mats + (w * 3 + l) * 8;
    m[0] =  c1 * cA;  m[1] = -c1 * sA;   // M00
    m[2] = -s1 * cB;  m[3] = -s1 * sB;   // M01
    m[4] =  s1 * cB;  m[5] = -s1 * sB;   // M10
    m[6] =  c1 * cA;  m[7] =  c1 * sA;   // M11
}

// ---------------------------------------------------------------------------
// Kernel B: per-pixel 3-wire evolution.
// One thread per pixel; each block covers 256 contiguous pixels of one image
// (1024 px/image -> 4 blocks/image), so the image base is wave-uniform and
// feeds the SADDR form of the CDNA5 async global->LDS DMA.
// ---------------------------------------------------------------------------
__global__ void __launch_bounds__(THREADS) qsim_kernel(
    const float* __restrict__ x,     // [bsz, 3, 32, 32]
    const float* __restrict__ mats,  // 72 floats (uniform -> s_load)
    float* __restrict__ out) {       // [bsz, 3, 32, 32]
    __shared__ float sx[3 * THREADS];

    const int tid = threadIdx.x;
    const int b   = blockIdx.x >> 2;                      // image index
    const int pix = ((blockIdx.x & 3) << 8) + tid;        // pixel in image
    const float* xb = x + (size_t)b * 3072;               // uniform per block

    // CDNA5 async DMA: stage the 3 channel values per thread into LDS.
    // Tracked by ASYNCcnt; each wave only consumes its own data, so a single
    // s_wait_asynccnt 0 (no barrier) is sufficient.
    uint32_t lds0 = (uint32_t)(uintptr_t)(&sx[0]);
#pragma unroll
    for (int w = 0; w < 3; ++w) {
        uint32_t lds = lds0 + (uint32_t)((w * THREADS + tid) * 4);
        uint32_t go  = (uint32_t)((w * 1024 + pix) * 4);
        asm volatile("global_load_async_to_lds_b32 %0, %1, %2"
                     :: "v"(lds), "v"(go), "s"(xb) : "memory");
    }
    asm volatile("s_wait_asynccnt 0" ::: "memory");

    float z[3];
#pragma unroll
    for (int w = 0; w < 3; ++w) {
        float xv = sx[w * THREADS + tid];
        // Hardware transcendental path (v_sin_f32 / v_cos_f32): ~1e-6 abs
        // error on the N(0,1)-ranged encoder angle, vs ~30 VALU ops saved
        // per call relative to the libm slow path. This is what moves the
        // kernel from transcendental-bound to memory-bound.
        float ha = 0.5f * xv;
        float cx = __cosf(ha);
        float sn = __sinf(ha);
        // v = RY(x)|0> = (cx, sn), real
        float v0r = cx, v0i = 0.f, v1r = sn, v1i = 0.f;
#pragma unroll
        for (int l = 0; l < 3; ++l) {
            const float* m = mats + (w * 3 + l) * 8;  // uniform scalar loads
            float m00r = m[0], m00i = m[1], m01r = m[2], m01i = m[3];
            float m10r = m[4], m10i = m[5], m11r = m[6], m11i = m[7];
            // v <- M_l v   (complex 2x2 * complex 2-vector)
            float n0r = m00r * v0r - m00i * v0i + m01r * v1r - m01i * v1i;
            float n0i = m00r * v0i + m00i * v0r + m01r * v1i + m01i * v1r;
            float n1r = m10r * v0r - m10i * v0i + m11r * v1r - m11i * v1i;
            float n1i = m10r * v0i + m10i * v0r + m11r * v1i + m11i * v1r;
            if (l < 2) {
                // v <- RY(x) v   (real rotation on complex components)
                v0r = cx * n0r - sn * n1r;  v0i = cx * n0i - sn * n1i;
                v1r = sn * n0r + cx * n1r;  v1i = sn * n0i + cx * n1i;
            } else {
                v0r = n0r; v0i = n0i; v1r = n1r; v1i = n1i;
            }
        }
        z[w] = (v0r * v0r + v0i * v0i) - (v1r * v1r + v1i * v1i);
    }

    float* ob = out + (size_t)b * 3072 + pix;   // coalesced plane stores
    ob[0]    = z[0];
    ob[1024] = z[1];
    ob[2048] = z[2];
}

// ---------------------------------------------------------------------------
extern "C" void kernel_launch(void* const* d_in, const int* in_sizes, int n_in,
                              void* d_out, int out_size, void* d_ws, size_t ws_size,
                              hipStream_t stream) {
    const float* x      = (const float*)d_in[0];   // [bsz,3,32,32]
    const float* params = (const float*)d_in[1];   // [3,9]
    float*       mats   = (float*)d_ws;            // 72 floats scratch
    float*       out    = (float*)d_out;

    const int bsz     = in_sizes[0] / 3072;        // 3*32*32 per image
    const int nblocks = bsz * 4;                   // 1024 px / 256 threads

    compose_gates<<<1, 32, 0, stream>>>(params, mats);
    qsim_kernel<<<nblocks, THREADS, 0, stream>>>(x, mats, out);
}